// SmallSMBlock_37039797960726
// MI455X (gfx1250) — compile-verified
//
#include <hip/hip_runtime.h>
#include <stdint.h>

#define NN 1024
#define TILE 32
#define HROWS 34
#define HCOLS 40            // cols tj-4 .. tj+35 (16B-aligned global rows)
#define SM_ELEMS (HROWS * HCOLS) // 1360
#define VEC4S (HROWS * (HCOLS / 4)) // 340 b128 transfers per tile

typedef __attribute__((ext_vector_type(2))) float v2f;
typedef __attribute__((ext_vector_type(8))) float v8f;

// ds_swizzle SWAPX16: group-of-32 mode, xor=0x10, or=0, and=0x1f -> 0x401F
__device__ __forceinline__ float swapx16(float v) {
    return __int_as_float(__builtin_amdgcn_ds_swizzle(__float_as_int(v), 0x401F));
}

// Issue staging of a 34x40 halo tile of one [N,N] plane into LDS.
// Interior tiles: CDNA5 async global->LDS b128 DMA (caller must s_wait_asynccnt before use).
// Edge tiles: bounds-checked scalar fallback (synchronous ds stores).
__device__ __forceinline__ void stage_issue(const float* __restrict__ src, float* sm,
                                            int ti, int tj, bool interior, int tid) {
    if (interior) {
        const float* base = src + (size_t)(ti - 1) * NN + (tj - 4);
#pragma unroll
        for (int k = 0; k < 2; ++k) {
            int e = tid + k * 256; // vec4 index, 340 total
            if (e < VEC4S) {
                int r = e / 10, v = e - r * 10;
                const float* gp = base + (size_t)r * NN + v * 4;
                unsigned lds_addr = (unsigned)(uintptr_t)(sm + (r * HCOLS + v * 4));
                asm volatile("global_load_async_to_lds_b128 %0, %1, off"
                             :: "v"(lds_addr), "v"(gp) : "memory");
            }
        }
    } else {
        for (int e = tid; e < SM_ELEMS; e += 256) {
            int r = e / HCOLS, c = e - r * HCOLS;
            int gr = ti + r - 1, gc = tj + c - 4;
            float v = 0.0f;
            if (gr >= 0 && gr < NN && gc >= 0 && gc < NN) v = src[(size_t)gr * NN + gc];
            sm[e] = v;
        }
    }
}

__device__ __forceinline__ void wait_async() {
    asm volatile("s_wait_asynccnt 0" ::: "memory");
}

__global__ __launch_bounds__(256) void smallsm_kernel(const float* __restrict__ image,
                                                      const float* __restrict__ x,
                                                      const float* __restrict__ weights,
                                                      const float* __restrict__ bias,
                                                      float* __restrict__ y) {
    __shared__ __attribute__((aligned(16))) float sm[2][SM_ELEMS];
    const int tid = threadIdx.x;
    const int lane = tid & 31;
    const int wave = tid >> 5;
    const int s = lane & 15;
    const bool hi = lane >= 16;
    const int ti = blockIdx.y * TILE;
    const int tj = blockIdx.x * TILE;
    const bool interior = (ti >= 1) && (ti + TILE <= NN - 1) && (tj >= 4) && (tj + 36 <= NN);

    // ---- A-matrix (weights) in V_WMMA_F32_16X16X4_F32 layout:
    // A[16x4] slice j: VGPR0 lane L: K=q0+(hi?2:0); VGPR1: K=q0+(hi?3:1); M=lane%16.
    v2f a[3];
#pragma unroll
    for (int j = 0; j < 3; ++j) {
        int qx = 4 * j + (hi ? 2 : 0);
        int qy = 4 * j + (hi ? 3 : 1);
        a[j].x = (s < 9 && qx < 9) ? weights[s * 9 + qx] : 0.0f;
        a[j].y = (s < 9 && qy < 9) ? weights[s * 9 + qy] : 0.0f;
    }
    // Per-lane B-matrix gather offsets into the LDS image tile (clamped when q invalid).
    int boffx[3], boffy[3];
    bool bokx[3], boky[3];
#pragma unroll
    for (int j = 0; j < 3; ++j) {
        int qx = 4 * j + (hi ? 2 : 0);
        int qy = 4 * j + (hi ? 3 : 1);
        bokx[j] = qx < 9;
        boky[j] = qy < 9;
        boffx[j] = bokx[j] ? (qx / 3) * HCOLS + (qx % 3) : 0;
        boffy[j] = boky[j] ? (qy / 3) * HCOLS + (qy % 3) : 0;
    }
    float bs[9];
#pragma unroll
    for (int p = 0; p < 9; ++p) bs[p] = bias[p];

    // ---- Stage image tile into buffer 0 ----
    stage_issue(image, sm[0], ti, tj, interior, tid);
    wait_async();
    __syncthreads();

    // ---- Prefetch x(b=0) into buffer 1 while WMMA computes the K field ----
    stage_issue(x, sm[1], ti, tj, interior, tid);

    // ---- Phase 1: K field via WMMA GEMM (K[p,pix] = sum_q W[p,q]*patch[q,pix]) ----
    const int R0 = wave * 4; // this wave owns tile rows R0..R0+3
    const float* img = sm[0];
    float kv[4][9]; // per-lane K coefficients: lane owns column `lane`
#pragma unroll
    for (int t = 0; t < 4; ++t) {
        int R = R0 + t;
        v8f dA = {}, dB = {};
#pragma unroll
        for (int j = 0; j < 3; ++j) {
            int baseA = R * HCOLS + s + 3;  // pixel group A: cols 0..15
            int baseB = baseA + 16;         // pixel group B: cols 16..31
            v2f bA, bB;
            float vax = img[baseA + boffx[j]], vay = img[baseA + boffy[j]];
            float vbx = img[baseB + boffx[j]], vby = img[baseB + boffy[j]];
            bA.x = bokx[j] ? vax : 0.0f;
            bA.y = boky[j] ? vay : 0.0f;
            bB.x = bokx[j] ? vbx : 0.0f;
            bB.y = boky[j] ? vby : 0.0f;
            dA = __builtin_amdgcn_wmma_f32_16x16x4_f32(false, a[j], false, bA, (short)0, dA,
                                                       false, false);
            dB = __builtin_amdgcn_wmma_f32_16x16x4_f32(false, a[j], false, bB, (short)0, dB,
                                                       false, false);
        }
        // Redistribute D so lane L owns all 9 K's of column L (SWAPX16 cross-lane moves).
#pragma unroll
        for (int p = 0; p < 8; ++p) {
            float swB = swapx16(dB[p]);
            kv[t][p] = (hi ? swB : dA[p]) + bs[p];
        }
        float swA0 = swapx16(dA[0]); // M=8 row lives in upper lanes of VGPR0
        kv[t][8] = (hi ? dB[0] : swA0) + bs[8];
    }

    wait_async();    // x(b=0) landed in buffer 1
    __syncthreads(); // and everyone is done reading the image buffer

    // ---- Phase 2: per-batch apply, double-buffered async pipeline ----
    const int C = lane; // owned column
    for (int b = 0; b < 16; ++b) {
        const float* cur = sm[(b + 1) & 1];
        float* nxt = sm[b & 1];
        if (b + 1 < 16) stage_issue(x + (size_t)(b + 1) * NN * NN, nxt, ti, tj, interior, tid);

        float xr[6][3];
#pragma unroll
        for (int rr = 0; rr < 6; ++rr)
#pragma unroll
            for (int cc = 0; cc < 3; ++cc) xr[rr][cc] = cur[(R0 + rr) * HCOLS + C + 3 + cc];
#pragma unroll
        for (int t = 0; t < 4; ++t) {
            float acc = 0.0f;
#pragma unroll
            for (int m = 0; m < 3; ++m)
#pragma unroll
                for (int n = 0; n < 3; ++n) acc = fmaf(kv[t][3 * m + n], xr[t + m][n], acc);
            y[((size_t)b * NN + (ti + R0 + t)) * NN + (tj + C)] = acc;
        }

        wait_async();    // next buffer fully landed (no-op on fallback path / last iter)
        __syncthreads(); // all reads of cur complete before it becomes nxt
    }
}

extern "C" void kernel_launch(void* const* d_in, const int* in_sizes, int n_in, void* d_out,
                              int out_size, void* d_ws, size_t ws_size, hipStream_t stream) {
    (void)in_sizes; (void)n_in; (void)out_size; (void)d_ws; (void)ws_size;
    const float* image = (const float*)d_in[0];
    const float* x = (const float*)d_in[1];
    const float* weights = (const float*)d_in[2];
    const float* bias = (const float*)d_in[3];
    float* y = (float*)d_out;
    dim3 grid(NN / TILE, NN / TILE);
    smallsm_kernel<<<grid, dim3(256), 0, stream>>>(image, x, weights, bias, y);
}